// HolyLossNetwork_13761075216391
// MI455X (gfx1250) — compile-verified
//
#include <hip/hip_runtime.h>
#include <math.h>
#include <stdint.h>

// ---------------- problem sizes ----------------
#define NEP    2048
#define NSHOT  32
#define IN_DIM 512
#define FEAT   1023
#define REPR   2048          // 2*(FEAT+1)
#define KCAT   8192          // 4 stacked 2048-wide segments for the fused KL GEMM

typedef __attribute__((ext_vector_type(2))) float v2f;
typedef __attribute__((ext_vector_type(8))) float v8f;
typedef int v4i __attribute__((vector_size(16)));   // 16B payload for async b128

// CDNA5 fp32 WMMA: D(16x16 f32) = A(16x4) * B(4x16) + C
static __device__ __forceinline__ v8f wmma_f32(v2f a, v2f b, v8f c) {
  return __builtin_amdgcn_wmma_f32_16x16x4_f32(false, a, false, b, (short)0, c,
                                               false, false);
}

// ---- gfx1250 async global->LDS staging (GLOBAL_LOAD_ASYNC_TO_LDS_B128,
// tracked by ASYNCcnt; completes in order per wave) ----
#if __has_builtin(__builtin_amdgcn_global_load_async_to_lds_b128) && \
    __has_builtin(__builtin_amdgcn_s_wait_asynccnt)
#define ASYNC_LDS 1
static __device__ __forceinline__ void async_copy16(const float* g, float* l) {
  __builtin_amdgcn_global_load_async_to_lds_b128((v4i*)g, (v4i*)l, 0, 0);
}
template <int N>
static __device__ __forceinline__ void async_wait_n() {
  __builtin_amdgcn_s_wait_asynccnt(N);
}
#else
static __device__ __forceinline__ void async_copy16(const float* g, float* l) {
  const float4 t = *(const float4*)g;
  *(float4*)l = t;
}
template <int N>
static __device__ __forceinline__ void async_wait_n() {}
#endif

static __device__ __forceinline__ float wave_sum(float v) {
  v += __shfl_xor(v, 16, 32);
  v += __shfl_xor(v, 8, 32);
  v += __shfl_xor(v, 4, 32);
  v += __shfl_xor(v, 2, 32);
  v += __shfl_xor(v, 1, 32);
  return v;
}
static __device__ __forceinline__ float wave_max(float v) {
  v = fmaxf(v, __shfl_xor(v, 16, 32));
  v = fmaxf(v, __shfl_xor(v, 8, 32));
  v = fmaxf(v, __shfl_xor(v, 4, 32));
  v = fmaxf(v, __shfl_xor(v, 2, 32));
  v = fmaxf(v, __shfl_xor(v, 1, 32));
  return v;
}

// ------------------------------------------------------------------
// K1: per-episode  phis = X_e(32x512) @ W_fe(512x1023), fused column
// mean/std(ddof=1) over the 32 shots -> z[e, 1+c] / z[e, 1025+c].
// Columns are CLAMPED (not predicated) so the inner loop is branch-free;
// the n=1023 tile lanes accumulate garbage that is never written out.
// ------------------------------------------------------------------
__global__ __launch_bounds__(256) void feat_stats_kernel(
    const float* __restrict__ X, const float* __restrict__ Wfe,
    const float* __restrict__ bfe, float* __restrict__ z) {
  const int e    = blockIdx.x;
  const int half = blockIdx.y;
  const int tid  = threadIdx.x;
  const int wave = tid >> 5, lane = tid & 31;
  const int nlane = lane & 15;
  const int koff  = (lane >> 4) * 2;        // K-split of A/B fragments
  const float* Xe = X + (size_t)e * (NSHOT * IN_DIM);
  const int colWave = half * 512 + wave * 64;

  int colc[4];
#pragma unroll
  for (int nt = 0; nt < 4; ++nt) {
    const int c = colWave + nt * 16 + nlane;
    colc[nt] = (c < FEAT) ? c : (FEAT - 1);
  }

  v8f acc[2][4] = {};
#pragma unroll 2
  for (int k0 = 0; k0 < IN_DIM; k0 += 4) {
    const int kb = k0 + koff;               // kb is even -> 8B aligned pair
    const v2f a0 = *(const v2f*)(Xe + (size_t)(nlane)      * IN_DIM + kb);
    const v2f a1 = *(const v2f*)(Xe + (size_t)(16 + nlane) * IN_DIM + kb);
    v2f b[4];
#pragma unroll
    for (int nt = 0; nt < 4; ++nt) {
      b[nt].x = Wfe[(size_t)kb * FEAT + colc[nt]];
      b[nt].y = Wfe[(size_t)(kb + 1) * FEAT + colc[nt]];
    }
#pragma unroll
    for (int nt = 0; nt < 4; ++nt) {
      acc[0][nt] = wmma_f32(a0, b[nt], acc[0][nt]);
      acc[1][nt] = wmma_f32(a1, b[nt], acc[1][nt]);
    }
  }

  // Column reduction over M (shots). C layout: VGPR r holds M=r (lanes 0-15)
  // and M=r+8 (lanes 16-31); N = lane&15.  std is shift-invariant -> fold
  // b_fe only into the mean.
#pragma unroll
  for (int nt = 0; nt < 4; ++nt) {
    float s = 0.0f, q = 0.0f;
#pragma unroll
    for (int r = 0; r < 8; ++r) {
      const float v0 = acc[0][nt][r];
      const float v1 = acc[1][nt][r];
      s += v0 + v1;
      q += v0 * v0 + v1 * v1;
    }
    s += __shfl_xor(s, 16, 32);
    q += __shfl_xor(q, 16, 32);
    const int col = colWave + nt * 16 + nlane;
    if (lane < 16 && col < FEAT) {
      const float mean = s * (1.0f / NSHOT) + bfe[col];
      float var = (q - s * s * (1.0f / NSHOT)) * (1.0f / (NSHOT - 1));
      var = fmaxf(var, 0.0f);
      z[(size_t)e * REPR + 1 + col]              = mean;
      z[(size_t)e * REPR + 1 + (FEAT + 1) + col] = sqrtf(var);
    }
  }
}

// K1b: targets column -> z[:,0] (mean), z[:,1024] (std)
__global__ __launch_bounds__(256) void target_stats_kernel(
    const float* __restrict__ T, float* __restrict__ z) {
  const int e = blockIdx.x * blockDim.x + threadIdx.x;
  if (e >= NEP) return;
  float s = 0.0f, q = 0.0f;
#pragma unroll
  for (int i = 0; i < NSHOT; ++i) {
    const float v = T[e * NSHOT + i];
    s += v; q += v * v;
  }
  const float mean = s * (1.0f / NSHOT);
  const float var =
      fmaxf((q - s * s * (1.0f / NSHOT)) * (1.0f / (NSHOT - 1)), 0.0f);
  z[(size_t)e * REPR]            = mean;
  z[(size_t)e * REPR + FEAT + 1] = sqrtf(var);
}

// ------------------------------------------------------------------
// K2: lv = clip(z @ W_lv + b_lv, -9, 1)  (NN WMMA GEMM 2048^3) with
// double-buffered async global->LDS staging.  Epilogue fuses exp() and
// writes the packed KL operand matrices:
//   tr side : Acat  row-major [2048 x 8192]: [0.5*(var+mu^2), mu, inv, -mu*inv]
//   te side : BcatT K-major  [8192 x 2048]: rows [inv, -mu*inv, 0.5*(var+mu^2),
//             mu] transposed, so the score GEMM is a plain NN GEMM.
// ------------------------------------------------------------------
__global__ __launch_bounds__(256) void gemm_lv_cat_kernel(
    const float* __restrict__ Z, const float* __restrict__ W,
    const float* __restrict__ bias, float* __restrict__ cat,
    float* __restrict__ lv_out, int te_side) {
  __shared__ float As[2][128][36];  // stride 36: 16B aligned + conflict-free
  __shared__ float Bs[2][32][132];
  const int tid = threadIdx.x;
  const int wave = tid >> 5, lane = tid & 31;
  const int wm = wave & 3, wn = wave >> 2;
  const int nlane = lane & 15;
  const int koff = (lane >> 4) * 2;
  const int hsel = (lane >> 4) * 8;
  const int rowBase = blockIdx.y * 128, colBase = blockIdx.x * 128;
  const int ar = tid >> 1, akc = (tid & 1) * 16;
  const int bkr = tid >> 3, bnc = (tid & 7) * 16;

  auto issue = [&](int kk, int ib) {      // 8 async b128 copies per thread
#pragma unroll
    for (int j = 0; j < 16; j += 4)
      async_copy16(Z + (size_t)(rowBase + ar) * REPR + kk + akc + j,
                   &As[ib][ar][akc + j]);
#pragma unroll
    for (int j = 0; j < 16; j += 4)
      async_copy16(W + (size_t)(kk + bkr) * REPR + colBase + bnc + j,
                   &Bs[ib][bkr][bnc + j]);
  };

  v8f acc[2][4] = {};
  issue(0, 0);
  for (int kk = 0; kk < REPR; kk += 32) {
    const int ib = (kk >> 5) & 1;
    if (kk + 32 < REPR) {                 // overlap next slab with compute
      issue(kk + 32, ib ^ 1);
      async_wait_n<8>();                  // slab kk done (in-order completion)
    } else {
      async_wait_n<0>();
    }
    __syncthreads();
#pragma unroll
    for (int k2 = 0; k2 < 32; k2 += 4) {
      const int kb = k2 + koff;
      v2f a[2], b[4];
#pragma unroll
      for (int mt = 0; mt < 2; ++mt)
        a[mt] = *(const v2f*)(&As[ib][wm * 32 + mt * 16 + nlane][kb]);
#pragma unroll
      for (int nt = 0; nt < 4; ++nt) {
        const int c = wn * 64 + nt * 16 + nlane;
        b[nt].x = Bs[ib][kb][c]; b[nt].y = Bs[ib][kb + 1][c];
      }
#pragma unroll
      for (int mt = 0; mt < 2; ++mt)
#pragma unroll
        for (int nt = 0; nt < 4; ++nt)
          acc[mt][nt] = wmma_f32(a[mt], b[nt], acc[mt][nt]);
    }
    __syncthreads();
  }

#pragma unroll
  for (int mt = 0; mt < 2; ++mt) {
#pragma unroll
    for (int nt = 0; nt < 4; ++nt) {
      const int n = colBase + wn * 64 + nt * 16 + nlane;
#pragma unroll
      for (int r = 0; r < 8; ++r) {
        const int m = rowBase + wm * 32 + mt * 16 + r + hsel;
        float lv = acc[mt][nt][r] + bias[n];
        lv = fminf(1.0f, fmaxf(-9.0f, lv));           // Hardtanh(-9,1)
        const float mu  = Z[(size_t)m * REPR + n];
        const float inv = __expf(-lv);
        const float var = __expf(lv);
        const float p   = 0.5f * (var + mu * mu);
        const float rr  = -mu * inv;
        if (!te_side) {
          const size_t cb = (size_t)m * KCAT;
          cat[cb + n]            = p;
          cat[cb + REPR + n]     = mu;
          cat[cb + 2 * REPR + n] = inv;
          cat[cb + 3 * REPR + n] = rr;
        } else {                      // transposed: row k = seg*REPR+n, col m
          cat[(size_t)(n)            * NEP + m] = inv;
          cat[(size_t)(REPR + n)     * NEP + m] = rr;
          cat[(size_t)(2 * REPR + n) * NEP + m] = p;
          cat[(size_t)(3 * REPR + n) * NEP + m] = mu;
          lv_out[(size_t)m * REPR + n] = lv;
        }
      }
    }
  }
}

// colbias[j] = sum_d lv_te[j,d] + 0.5 * sum_d mu^2 * exp(-lv)   (only the
// column-varying part of score survives log_softmax over axis=1)
__global__ __launch_bounds__(256) void colbias_kernel(
    const float* __restrict__ lv_te, const float* __restrict__ z_te,
    float* __restrict__ cb) {
  const int j = blockIdx.x;
  const int tid = threadIdx.x;
  float s = 0.0f;
  for (int d = tid; d < REPR; d += 256) {
    const float lv = lv_te[(size_t)j * REPR + d];
    const float mu = z_te[(size_t)j * REPR + d];
    s += lv + 0.5f * mu * mu * __expf(-lv);
  }
  __shared__ float red[8];
  s = wave_sum(s);
  if ((tid & 31) == 0) red[tid >> 5] = s;
  __syncthreads();
  if (tid == 0) {
    float t = 0.0f;
#pragma unroll
    for (int w = 0; w < 8; ++w) t += red[w];
    cb[j] = t;
  }
}

// ------------------------------------------------------------------
// K3: score = Acat(2048x8192) @ BcatT(8192x2048) + colbias[j]  (NN GEMM),
// double-buffered async staging.
// ------------------------------------------------------------------
__global__ __launch_bounds__(256) void gemm_score_kernel(
    const float* __restrict__ A, const float* __restrict__ BT,
    const float* __restrict__ colbias, float* __restrict__ score) {
  __shared__ float As[2][128][36];
  __shared__ float Bs[2][32][132];
  const int tid = threadIdx.x;
  const int wave = tid >> 5, lane = tid & 31;
  const int wm = wave & 3, wn = wave >> 2;
  const int nlane = lane & 15;
  const int koff = (lane >> 4) * 2;
  const int hsel = (lane >> 4) * 8;
  const int rowBase = blockIdx.y * 128, colBase = blockIdx.x * 128;
  const int ar = tid >> 1, akc = (tid & 1) * 16;
  const int bkr = tid >> 3, bnc = (tid & 7) * 16;

  auto issue = [&](int kk, int ib) {
#pragma unroll
    for (int j = 0; j < 16; j += 4)
      async_copy16(A + (size_t)(rowBase + ar) * KCAT + kk + akc + j,
                   &As[ib][ar][akc + j]);
#pragma unroll
    for (int j = 0; j < 16; j += 4)
      async_copy16(BT + (size_t)(kk + bkr) * NEP + colBase + bnc + j,
                   &Bs[ib][bkr][bnc + j]);
  };

  v8f acc[2][4] = {};
  issue(0, 0);
  for (int kk = 0; kk < KCAT; kk += 32) {
    const int ib = (kk >> 5) & 1;
    if (kk + 64 < KCAT)                   // L2 prefetch two slabs ahead
      __builtin_prefetch(A + (size_t)(rowBase + ar) * KCAT + kk + 64 + akc, 0, 0);
    if (kk + 32 < KCAT) {
      issue(kk + 32, ib ^ 1);
      async_wait_n<8>();
    } else {
      async_wait_n<0>();
    }
    __syncthreads();
#pragma unroll
    for (int k2 = 0; k2 < 32; k2 += 4) {
      const int kb = k2 + koff;
      v2f a[2], b[4];
#pragma unroll
      for (int mt = 0; mt < 2; ++mt)
        a[mt] = *(const v2f*)(&As[ib][wm * 32 + mt * 16 + nlane][kb]);
#pragma unroll
      for (int nt = 0; nt < 4; ++nt) {
        const int c = wn * 64 + nt * 16 + nlane;
        b[nt].x = Bs[ib][kb][c]; b[nt].y = Bs[ib][kb + 1][c];
      }
#pragma unroll
      for (int mt = 0; mt < 2; ++mt)
#pragma unroll
        for (int nt = 0; nt < 4; ++nt)
          acc[mt][nt] = wmma_f32(a[mt], b[nt], acc[mt][nt]);
    }
    __syncthreads();
  }

#pragma unroll
  for (int mt = 0; mt < 2; ++mt) {
#pragma unroll
    for (int nt = 0; nt < 4; ++nt) {
      const int n = colBase + wn * 64 + nt * 16 + nlane;
      const float cbn = colbias[n];
#pragma unroll
      for (int r = 0; r < 8; ++r) {
        const int m = rowBase + wm * 32 + mt * 16 + r + hsel;
        score[(size_t)m * REPR + n] = acc[mt][nt][r] + cbn;
      }
    }
  }
}

// K4: probs[i,:] = log_softmax(-score[i,:])
__global__ __launch_bounds__(256) void logsoftmax_kernel(
    const float* __restrict__ score, float* __restrict__ probs) {
  const int i = blockIdx.x;
  const int tid = threadIdx.x;
  float xv[8];
  float mx = -3.402823466e38f;
#pragma unroll
  for (int t = 0; t < 8; ++t) {
    xv[t] = -score[(size_t)i * REPR + tid + t * 256];
    mx = fmaxf(mx, xv[t]);
  }
  __shared__ float red[8];
  mx = wave_max(mx);
  if ((tid & 31) == 0) red[tid >> 5] = mx;
  __syncthreads();
  float bm = red[0];
#pragma unroll
  for (int w = 1; w < 8; ++w) bm = fmaxf(bm, red[w]);
  __syncthreads();
  float s = 0.0f;
#pragma unroll
  for (int t = 0; t < 8; ++t) s += __expf(xv[t] - bm);
  s = wave_sum(s);
  if ((tid & 31) == 0) red[tid >> 5] = s;
  __syncthreads();
  float tot = 0.0f;
#pragma unroll
  for (int w = 0; w < 8; ++w) tot += red[w];
  const float lse = bm + logf(tot);
#pragma unroll
  for (int t = 0; t < 8; ++t)
    probs[(size_t)i * REPR + tid + t * 256] = xv[t] - lse;
}

__global__ void classes_kernel(int* __restrict__ cls) {
  const int j = blockIdx.x * blockDim.x + threadIdx.x;
  if (j < NEP) cls[j] = j;
}

// ------------------------------------------------------------------
extern "C" void kernel_launch(void* const* d_in, const int* in_sizes, int n_in,
                              void* d_out, int out_size, void* d_ws,
                              size_t ws_size, hipStream_t stream) {
  const float* in_tr = (const float*)d_in[0];
  const float* tg_tr = (const float*)d_in[1];
  const float* in_te = (const float*)d_in[2];
  const float* tg_te = (const float*)d_in[3];
  const float* Wfe   = (const float*)d_in[4];
  const float* bfe   = (const float*)d_in[5];
  const float* Wlv   = (const float*)d_in[6];
  const float* blv   = (const float*)d_in[7];

  float* ws = (float*)d_ws;
  const size_t Z = (size_t)REPR * REPR;     // 4M floats
  float* z_tr  = ws;                        // [2048,2048]
  float* z_te  = ws + Z;                    // [2048,2048]
  float* Acat  = ws + 2 * Z;                // [2048,8192] row-major
  float* BcatT = ws + 6 * Z;                // [8192,2048] K-major
  float* lv_te = ws + 10 * Z;               // [2048,2048]
  float* scre  = ws + 11 * Z;               // [2048,2048]
  float* cbias = ws + 12 * Z;               // [2048]
  float* probs = (float*)d_out;
  int*   cls   = (int*)((float*)d_out + Z);

  feat_stats_kernel<<<dim3(NEP, 2), 256, 0, stream>>>(in_tr, Wfe, bfe, z_tr);
  feat_stats_kernel<<<dim3(NEP, 2), 256, 0, stream>>>(in_te, Wfe, bfe, z_te);
  target_stats_kernel<<<dim3((NEP + 255) / 256), 256, 0, stream>>>(tg_tr, z_tr);
  target_stats_kernel<<<dim3((NEP + 255) / 256), 256, 0, stream>>>(tg_te, z_te);
  gemm_lv_cat_kernel<<<dim3(16, 16), 256, 0, stream>>>(z_tr, Wlv, blv, Acat,
                                                       nullptr, 0);
  gemm_lv_cat_kernel<<<dim3(16, 16), 256, 0, stream>>>(z_te, Wlv, blv, BcatT,
                                                       lv_te, 1);
  colbias_kernel<<<dim3(NEP), 256, 0, stream>>>(lv_te, z_te, cbias);
  gemm_score_kernel<<<dim3(16, 16), 256, 0, stream>>>(Acat, BcatT, cbias, scre);
  logsoftmax_kernel<<<dim3(NEP), 256, 0, stream>>>(scre, probs);
  classes_kernel<<<dim3((NEP + 255) / 256), 256, 0, stream>>>(cls);

  (void)in_sizes; (void)n_in; (void)out_size; (void)ws_size;
}